// LinearScaling_17257178595753
// MI455X (gfx1250) — compile-verified
//
#include <hip/hip_runtime.h>

// LinearScaling: out[r, c] = x[r, c] * w[c]
//   x: [8192, 4096] fp32, w: [4096] fp32, out: [8192, 4096] fp32
//
// Pure streaming op: 268 MB of traffic vs 33.5 MFLOP -> memory-bound by >30x
// on MI455X (23.3 TB/s => ~11.5 us floor). Strategy: b128 nontemporal
// loads/stores for the 256 MB x/out streams (gfx1250 TH=NT cache policy, keeps
// L2/WGP$ clean), regular load for the 16 KiB weight vector so it stays
// cache-resident, and 16 independent in-flight b128 loads per thread for MLP.

constexpr int SIZE_IN        = 4096;
constexpr int ROWS           = 8192;
constexpr int THREADS        = 256;               // 8 wave32s per block
constexpr int COLS_V4        = SIZE_IN / 4;       // 1024 float4 per row
constexpr int SEGS           = COLS_V4 / THREADS; // 4 column segments per row
constexpr int ROWS_PER_BLOCK = 16;                // weight reuse + 16-deep MLP

typedef float v4f __attribute__((ext_vector_type(4)));

__global__ __launch_bounds__(THREADS)
void LinearScaling_scale_cols(const float* __restrict__ x,
                              const float* __restrict__ w,
                              float* __restrict__ out) {
    // Column slot (in float4 units) this thread owns for all its rows.
    const int cv4  = blockIdx.x * THREADS + threadIdx.x;   // [0, 1024)
    const int row0 = blockIdx.y * ROWS_PER_BLOCK;

    // Weight vector: tiny (16 KiB total), reused by every row in the grid.
    // Load with default (regular temporal) policy so it stays hot in cache.
    const v4f wv = ((const v4f*)w)[cv4];

    const v4f* __restrict__ xin = (const v4f*)x   + (size_t)row0 * COLS_V4 + cv4;
    v4f*       __restrict__ o   = (v4f*)out       + (size_t)row0 * COLS_V4 + cv4;

    // Issue all 16 row loads up front (independent addresses -> 16 outstanding
    // b128 nontemporal loads per thread; compiler batches the s_wait_loadcnt).
    v4f r[ROWS_PER_BLOCK];
#pragma unroll
    for (int i = 0; i < ROWS_PER_BLOCK; ++i) {
        r[i] = __builtin_nontemporal_load(&xin[(size_t)i * COLS_V4]);
    }

    // Scale and stream out, again nontemporal: this data will not be re-read.
#pragma unroll
    for (int i = 0; i < ROWS_PER_BLOCK; ++i) {
        v4f v = r[i] * wv;
        __builtin_nontemporal_store(v, &o[(size_t)i * COLS_V4]);
    }
}

extern "C" void kernel_launch(void* const* d_in, const int* in_sizes, int n_in,
                              void* d_out, int out_size, void* d_ws, size_t ws_size,
                              hipStream_t stream) {
    const float* x = (const float*)d_in[0];   // [ROWS * SIZE_IN]
    const float* w = (const float*)d_in[1];   // [SIZE_IN]
    float* out     = (float*)d_out;           // [ROWS * SIZE_IN]

    (void)in_sizes; (void)n_in; (void)out_size; (void)d_ws; (void)ws_size;

    dim3 grid(SEGS, ROWS / ROWS_PER_BLOCK);   // (4, 512) = 2048 blocks
    LinearScaling_scale_cols<<<grid, THREADS, 0, stream>>>(x, w, out);
}